// TemporalTransformerHawkesGraphModel_26371099197734
// MI455X (gfx1250) — compile-verified
//
#include <hip/hip_runtime.h>
#include <hip/hip_bf16.h>
#include <math.h>

// ---------------------------------------------------------------------------
// Problem constants (reference: bs=1024, L=512, d=200, N_ENT=20000)
// ---------------------------------------------------------------------------
#define N_ENT_C   20000
#define PAD_E     19999
#define L_C       512
#define D_C       200      // d
#define K_C       600      // 3*d
#define HSIZE     1024     // LDS hash slots (>= 2x max distinct local entities)
#define HMASK     (HSIZE - 1)
#define BITWORDS  ((N_ENT_C + 31) / 32)   // 625

typedef __attribute__((ext_vector_type(2))) float v2f;
typedef __attribute__((ext_vector_type(8))) float v8f;

__device__ __forceinline__ float softplus_b10(float v) {
    // softplus(10*v)/10, overflow-safe
    float x = 10.0f * v;
    return (fmaxf(x, 0.0f) + log1pf(__expf(-fabsf(x)))) * 0.1f;
}

// ---------------------------------------------------------------------------
// Kernel 1: inten_raw[b,n] = sum_k x[b,k] * W[n,k],  x = [qe | seq | qr]
// M=bs, K=600, N=200.  V_WMMA_F32_16X16X4_F32, one wave per 16x16 C tile.
// A 16x4 layout (ISA 7.12.2): lanes0-15 M=lane, V0=K0,V1=K1; lanes16-31 K2/K3.
// B 4x16 symmetric: VGPRj = K(2*khalf + j), N = lane&15.
// C 16x16: VGPRj -> M = j + 8*(lane>>4), N = lane&15.
//
// N-tail handling: B column n only contributes to C column n, so tail lanes
// (ncol >= 200) simply read W row 0 (clamped, in-bounds) and their C columns
// are discarded at store time. No masking, no divergence in the K loop.
// ---------------------------------------------------------------------------
__global__ __launch_bounds__(32) void inten_gemm_wmma(
    const float* __restrict__ qe, const float* __restrict__ so,
    const float* __restrict__ qr, const float* __restrict__ W,
    float* __restrict__ out)
{
    const int lane  = threadIdx.x;
    const int mrow  = lane & 15;
    const int khalf = lane >> 4;          // 0 or 1
    const int m0    = blockIdx.x * 16;
    const int n0    = blockIdx.y * 16;
    const int ncol  = n0 + mrow;
    const int nclmp = (ncol < D_C) ? ncol : 0;

    // B pointer walks the full K range continuously across the 3 regions.
    const float* wp = W + (size_t)nclmp * K_C + 2 * khalf;
    const size_t arow_off = (size_t)(m0 + mrow) * D_C + 2 * khalf;

    v8f acc = {};

    const float* ap;

#define WMMA_K_REGION(SRC)                                                   \
    ap = (SRC) + arow_off;                                                   \
    _Pragma("unroll 5")                                                      \
    for (int kt = 0; kt < D_C / 4; ++kt) {                                   \
        v2f a;  a.x  = ap[0]; a.y  = ap[1];                                  \
        v2f bf; bf.x = wp[0]; bf.y = wp[1];                                  \
        acc = __builtin_amdgcn_wmma_f32_16x16x4_f32(                         \
            false, a, false, bf, (short)0, acc, false, false);               \
        ap += 4; wp += 4;                                                    \
    }

    WMMA_K_REGION(qe)
    WMMA_K_REGION(so)
    WMMA_K_REGION(qr)
#undef WMMA_K_REGION

    if (ncol < D_C) {
        #pragma unroll
        for (int j = 0; j < 8; ++j) {
            const int cm = m0 + j + 8 * khalf;
            out[(size_t)cm * D_C + ncol] = acc[j];
        }
    }
}

// ---------------------------------------------------------------------------
// Kernel 2: one block per batch row b. Fully fused:
//  (a) l_int = local_h[b] @ inten_raw[b], scale, +1, softplus
//  (b) scatter (sum,count) per entity into LDS hash + bitmap
//  (c) stream global_intes[b]: pooled value per entity, online logsumexp,
//      sum of logits, answer logit -> 4 partials per row
// ---------------------------------------------------------------------------
__global__ __launch_bounds__(256) void row_fused(
    const float* __restrict__ inten, const float* __restrict__ local_h,
    const float* __restrict__ g_int, const int* __restrict__ local_type,
    const int* __restrict__ answers, float* __restrict__ partials)
{
    __shared__ __align__(16) float s_int[D_C];
    __shared__ int      s_key[HSIZE];
    __shared__ float    s_sum[HSIZE];
    __shared__ float    s_cnt[HSIZE];
    __shared__ unsigned s_bits[BITWORDS];
    __shared__ float    r_a[256], r_m[256], r_s[256];
    __shared__ float    s_alogit;

    const int b = blockIdx.x;
    const int t = threadIdx.x;

    for (int i = t; i < HSIZE; i += 256) { s_key[i] = -1; s_sum[i] = 0.0f; s_cnt[i] = 0.0f; }
    for (int i = t; i < BITWORDS; i += 256) s_bits[i] = 0u;
    for (int i = t; i < D_C; i += 256) s_int[i] = inten[(size_t)b * D_C + i];
    if (t == 0) s_alogit = 0.0f;
    __syncthreads();

    const int ans = answers[b];

    // ---- (a)+(b): local rows (2 per thread) ----
    for (int l = t; l < L_C; l += 256) {
        const float4* row = (const float4*)(local_h + ((size_t)b * L_C + l) * D_C);
        if (l + 256 < L_C)
            __builtin_prefetch(local_h + ((size_t)b * L_C + l + 256) * D_C, 0, 0);

        const float4* si = (const float4*)s_int;
        float acc = 0.0f;
        #pragma unroll 10
        for (int k4 = 0; k4 < D_C / 4; ++k4) {
            float4 v = row[k4];
            float4 w = si[k4];
            acc = fmaf(v.x, w.x, acc);
            acc = fmaf(v.y, w.y, acc);
            acc = fmaf(v.z, w.z, acc);
            acc = fmaf(v.w, w.w, acc);
        }
        const int e = local_type[(size_t)b * L_C + l];
        const float scale = (e != PAD_E) ? 1.2f : 0.2f;
        const float sp = softplus_b10(acc * scale + 1.0f);

        atomicOr(&s_bits[e >> 5], 1u << (e & 31));
        int h = e & HMASK;
        for (;;) {
            int old = atomicCAS(&s_key[h], -1, e);
            if (old == -1 || old == e) {
                atomicAdd(&s_sum[h], sp);
                atomicAdd(&s_cnt[h], 1.0f);
                break;
            }
            h = (h + 1) & HMASK;
        }
    }
    __syncthreads();

    // ---- (c): stream global part, online logsumexp ----
    float sumL = 0.0f, m = -INFINITY, s = 0.0f;
    const float* grow = g_int + (size_t)b * N_ENT_C;
    for (int e = t; e < N_ENT_C; e += 256) {
        const float g = grow[e];
        float p = softplus_b10(g * ((e != PAD_E) ? 1.2f : 0.2f));
        if ((s_bits[e >> 5] >> (e & 31)) & 1u) {
            int h = e & HMASK;
            while (s_key[h] != e) h = (h + 1) & HMASK;
            p = (p + s_sum[h]) / (1.0f + s_cnt[h]);
        }
        if (e != PAD_E) {                       // logits drop the PAD column
            sumL += p;
            if (p > m) { s = s * __expf(m - p) + 1.0f; m = p; }
            else       { s += __expf(p - m); }
            if (e == ans) s_alogit = p;
        }
    }

    r_a[t] = sumL; r_m[t] = m; r_s[t] = s;
    __syncthreads();
    for (int off = 128; off > 0; off >>= 1) {
        if (t < off) {
            const float m1 = r_m[t], s1 = r_s[t];
            const float m2 = r_m[t + off], s2 = r_s[t + off];
            const float mm = fmaxf(m1, m2);
            r_m[t] = mm;
            r_s[t] = s1 * __expf(m1 - mm) + s2 * __expf(m2 - mm);
            r_a[t] = r_a[t] + r_a[t + off];
        }
        __syncthreads();
    }
    if (t == 0) {
        partials[4 * (size_t)b + 0] = r_a[0];
        partials[4 * (size_t)b + 1] = r_m[0];
        partials[4 * (size_t)b + 2] = r_s[0];
        partials[4 * (size_t)b + 3] = s_alogit;
    }
}

// ---------------------------------------------------------------------------
// Kernel 3: reduce 1024 row partials -> scalar loss
// ---------------------------------------------------------------------------
__global__ __launch_bounds__(256) void finalize_loss(
    const float* __restrict__ partials, float* __restrict__ out, int bs)
{
    __shared__ float r_sm[256], r_nl[256];
    const int t = threadIdx.x;
    const float c = (float)(N_ENT_C - 1);   // 19999

    float sm = 0.0f, nl = 0.0f;
    for (int b = t; b < bs; b += 256) {
        const float sumL = partials[4 * (size_t)b + 0];
        const float m    = partials[4 * (size_t)b + 1];
        const float s    = partials[4 * (size_t)b + 2];
        const float al   = partials[4 * (size_t)b + 3];
        const float logZ = m + __logf(s);
        sm += c * logZ - sumL;   // -sum(log_preds)
        nl += logZ - al;         // -log_pred[answer]
    }
    r_sm[t] = sm; r_nl[t] = nl;
    __syncthreads();
    for (int off = 128; off > 0; off >>= 1) {
        if (t < off) { r_sm[t] += r_sm[t + off]; r_nl[t] += r_nl[t + off]; }
        __syncthreads();
    }
    if (t == 0) {
        const float inv_bs = 1.0f / (float)bs;
        const float smooth = r_sm[0] * inv_bs;
        const float nll    = r_nl[0] * inv_bs;
        out[0] = smooth * (0.2f / c) + 0.8f * nll;
    }
}

// ---------------------------------------------------------------------------
extern "C" void kernel_launch(void* const* d_in, const int* in_sizes, int n_in,
                              void* d_out, int out_size, void* d_ws, size_t ws_size,
                              hipStream_t stream) {
    const float* qe = (const float*)d_in[0];   // query_ent_embeds (bs,200)
    const float* qr = (const float*)d_in[1];   // query_rel_embeds (bs,200)
    const float* so = (const float*)d_in[2];   // seq_output       (bs,200)
    const float* W  = (const float*)d_in[3];   // W_inten          (200,600)
    const float* lh = (const float*)d_in[4];   // local_h          (bs,512,200)
    const float* gi = (const float*)d_in[5];   // global_intes     (bs,20000)
    const int*   lt = (const int*)d_in[6];     // local_type       (bs,512)
    const int*   an = (const int*)d_in[7];     // answers          (bs,)

    const int bs = in_sizes[0] / D_C;          // 1024

    float* inten    = (float*)d_ws;                      // bs*200 floats
    float* partials = inten + (size_t)bs * D_C;          // bs*4 floats

    dim3 g1(bs / 16, (D_C + 15) / 16);                   // 64 x 13 tiles
    inten_gemm_wmma<<<g1, 32, 0, stream>>>(qe, so, qr, W, inten);
    row_fused<<<bs, 256, 0, stream>>>(inten, lh, gi, lt, an, partials);
    finalize_loss<<<1, 256, 0, stream>>>(partials, (float*)d_out, bs);
}